// DCE_48249662603419
// MI455X (gfx1250) — compile-verified
//
#include <hip/hip_runtime.h>

// ---- types for WMMA fragments (wave32, gfx1250) ----
typedef float v2f __attribute__((ext_vector_type(2)));   // A/B frag of V_WMMA_F32_16X16X4_F32 (64 f32 / 32 lanes)
typedef float v8f __attribute__((ext_vector_type(8)));   // 16x16 f32 C/D (256 f32 / 32 lanes)

#define HW320   320
#define PIX     (HW320 * HW320)       // 102400
#define NSAMP   50
#define NTILE16 (HW320 / 16)          // 20
#define MB64    (HW320 / 64)          // 5  (M blocked 4x16)
#define TILES_PER_IMG (MB64 * NTILE16)            // 100
#define TOTAL_TILES   (NSAMP * TILES_PER_IMG)     // 5000
#define ROWSTEP (16 * HW320)          // 5120 elements between M-subtile rows
#define PI_F    3.14159265358979323846f

// ---------------------------------------------------------------------------
// K0: coefficients.  K_time = cumsum(aifci)*0.1 (sequential f32, matches ref);
//     idx[s] = argmin_j |t_samp[j]-sample_time[s]| (strict < keeps first).
// ---------------------------------------------------------------------------
__global__ void k_coef(const float* __restrict__ aifci,
                       const float* __restrict__ t_samp,
                       const float* __restrict__ stime,
                       int T, int nsamp,
                       float* __restrict__ coefK, float* __restrict__ coefA)
{
    __shared__ float cum[608];
    if (threadIdx.x == 0) {
        float acc = 0.0f;
        for (int j = 0; j < T; ++j) { acc += aifci[j]; cum[j] = acc; }
    }
    __syncthreads();
    int s = threadIdx.x;
    if (s < nsamp) {
        float st = stime[s];
        int best = 0;
        float bd = fabsf(t_samp[0] - st);
        for (int j = 1; j < T; ++j) {
            float d = fabsf(t_samp[j] - st);
            if (d < bd) { bd = d; best = j; }
        }
        coefK[s] = cum[best] * 0.1f;   // STEP
        coefA[s] = aifci[best];
    }
}

// ---------------------------------------------------------------------------
// K1: shifted-DFT matrix  G[k,n] = (-1)^(k+n) * exp(-2*pi*i*k*n/320)
//     (fftshift o FFT o fftshift == multiply by G;  G symmetric).
//     Angle reduced via (k*n mod 320) so float sincos is accurate.
// ---------------------------------------------------------------------------
__global__ void k_dftmat(float* __restrict__ Gr, float* __restrict__ Gi)
{
    int i = blockIdx.x * 256 + threadIdx.x;      // 102400 total
    int k = i / HW320, n = i - k * HW320;
    int r = (k * n) % HW320;
    float th = -(2.0f * PI_F / (float)HW320) * (float)r;
    float sn, cs; sincosf(th, &sn, &cs);
    float sgn = ((k + n) & 1) ? -1.0f : 1.0f;
    Gr[i] = sgn * cs;
    Gi[i] = sgn * sn;
}

// ---------------------------------------------------------------------------
// K2: pointwise DCE signal model -> PLANAR sig staged in d_out:
//     sigR = out[0 .. 5.12M), sigI = out[5.12M .. 10.24M)
// ---------------------------------------------------------------------------
__global__ void k_sig(const float* __restrict__ x,       // [2,320,320,2]
                      const float* __restrict__ coefK,
                      const float* __restrict__ coefA,
                      float* __restrict__ sigR,
                      float* __restrict__ sigI)
{
    int i = blockIdx.x * 256 + threadIdx.x;              // 0 .. 50*102400-1
    int s = i / PIX;
    int p = i - s * PIX;

    const float TRc = 0.00487f, R1c = 1.0f, R1CA = 4.3f;
    const float fa  = 10.0f * PI_F / 180.0f;
    float cfa = cosf(fa);
    float E1  = expf(-TRc * R1c);
    float M0t = 100.0f * (1.0f - cfa * E1) / (1.0f - E1);          // M0*sin(fa)
    float Mst = M0t * (1.0f - E1) / (1.0f - E1 * cfa);             // M_steady
    float off = 100.0f - Mst;

    float cK = coefK[s], cA = coefA[s];
    float x0r = x[p * 2],            x0i = x[p * 2 + 1];
    float x1r = x[2 * PIX + p * 2],  x1i = x[2 * PIX + p * 2 + 1];
    float CAr = cK * x0r + cA * x1r;
    float CAi = cK * x0i + cA * x1i;

    float zr = -TRc * (R1c + R1CA * CAr);
    float zi = -TRc * (R1CA * CAi);
    float ex = expf(zr);
    float szi, czi; sincosf(zi, &szi, &czi);
    float Er = ex * czi, Ei = ex * szi;                   // complex exp

    float nr = 1.0f - Er,        ni = -Ei;                // 1 - E
    float dr = 1.0f - cfa * Er,  di = -cfa * Ei;          // 1 - cos(fa)*E
    float inv = 1.0f / (dr * dr + di * di);
    float qr = (nr * dr + ni * di) * inv;
    float qi = (ni * dr - nr * di) * inv;

    sigR[i] = M0t * qr + off;
    sigI[i] = M0t * qi;
}

// ---------------------------------------------------------------------------
// Complex tile GEMM core on V_WMMA_F32_16X16X4_F32, 64(M) x 16(N) per wave,
// software-pipelined (double-buffered fragments => loads for step k+1 are in
// flight while step k's 16 WMMAs execute; one wait boundary per k-step).
// All 8 A loads of a k-step share 2 dynamic addresses; the 4 row offsets
// (16*320 elements = 20480 B apart) fold into the 24-bit instruction offset.
// Fragment layouts (ISA 7.12.2, f32, wave32):
//   A 16x4 : lanes 0-15 M=lane, vgpr0/1 = K0/K1 ; lanes 16-31 = K2/K3
//   B 4x16 : lanes 0-15 N=lane, vgpr0/1 = K0/K1 ; lanes 16-31 = K2/K3
//   C 16x16: vgpr v, lanes 0-15 -> (M=v, N=lane); lanes 16-31 -> (M=v+8)
// Complex product per k-step (f32 WMMA has no A/B NEG, so negate B once):
//   Cr[j] += ar[j]*br + ai[j]*(-bi) ; Ci[j] += ar[j]*bi + ai[j]*br
// ---------------------------------------------------------------------------
static __device__ __forceinline__ v8f wmma4(v2f a, v2f b, v8f c) {
    return __builtin_amdgcn_wmma_f32_16x16x4_f32(false, a, false, b, (short)0, c, false, false);
}

struct CFrag {
    v2f br, bi;
    v2f ar0, ai0, ar1, ai1, ar2, ai2, ar3, ai3;
};

static __device__ __forceinline__ CFrag
ld_frag(const float* __restrict__ Ar, const float* __restrict__ Ai, unsigned r0,
        const float* __restrict__ Br, const float* __restrict__ Bi, unsigned boff,
        unsigned kk)
{
    const float* apr = Ar + r0 + kk;
    const float* api = Ai + r0 + kk;
    const float* bpr = Br + boff + kk * HW320;
    const float* bpi = Bi + boff + kk * HW320;
    CFrag f;
    f.br  = (v2f){ bpr[0], bpr[HW320] };                 // offset:0 / offset:1280
    f.bi  = (v2f){ bpi[0], bpi[HW320] };
    f.ar0 = *(const v2f*)(apr);                f.ai0 = *(const v2f*)(api);
    f.ar1 = *(const v2f*)(apr + ROWSTEP);      f.ai1 = *(const v2f*)(api + ROWSTEP);
    f.ar2 = *(const v2f*)(apr + 2 * ROWSTEP);  f.ai2 = *(const v2f*)(api + 2 * ROWSTEP);
    f.ar3 = *(const v2f*)(apr + 3 * ROWSTEP);  f.ai3 = *(const v2f*)(api + 3 * ROWSTEP);
    return f;
}

static __device__ __forceinline__ void
mma_frag(const CFrag& f, v8f accR[4], v8f accI[4])
{
    v2f nbi = (v2f){ -f.bi.x, -f.bi.y };                 // shared by 4 M-subtiles
    accR[0] = wmma4(f.ar0, f.br, accR[0]);  accR[0] = wmma4(f.ai0, nbi,  accR[0]);
    accI[0] = wmma4(f.ar0, f.bi, accI[0]);  accI[0] = wmma4(f.ai0, f.br, accI[0]);
    accR[1] = wmma4(f.ar1, f.br, accR[1]);  accR[1] = wmma4(f.ai1, nbi,  accR[1]);
    accI[1] = wmma4(f.ar1, f.bi, accI[1]);  accI[1] = wmma4(f.ai1, f.br, accI[1]);
    accR[2] = wmma4(f.ar2, f.br, accR[2]);  accR[2] = wmma4(f.ai2, nbi,  accR[2]);
    accI[2] = wmma4(f.ar2, f.bi, accI[2]);  accI[2] = wmma4(f.ai2, f.br, accI[2]);
    accR[3] = wmma4(f.ar3, f.br, accR[3]);  accR[3] = wmma4(f.ai3, nbi,  accR[3]);
    accI[3] = wmma4(f.ar3, f.bi, accI[3]);  accI[3] = wmma4(f.ai3, f.br, accI[3]);
}

static __device__ __forceinline__ void
cgemm_mainloop(const float* __restrict__ Ar, const float* __restrict__ Ai, unsigned r0,
               const float* __restrict__ Br, const float* __restrict__ Bi, unsigned boff,
               unsigned koff, v8f accR[4], v8f accI[4])
{
    // 80 k-steps of 4; peeled + double-buffered (prefetch stays in-bounds).
    CFrag f0 = ld_frag(Ar, Ai, r0, Br, Bi, boff, koff + 0u);
    for (int k = 0; k + 8 < HW320; k += 8) {
        CFrag f1 = ld_frag(Ar, Ai, r0, Br, Bi, boff, koff + (unsigned)k + 4u);
        mma_frag(f0, accR, accI);
        f0 = ld_frag(Ar, Ai, r0, Br, Bi, boff, koff + (unsigned)k + 8u);
        mma_frag(f1, accR, accI);
    }
    CFrag f1 = ld_frag(Ar, Ai, r0, Br, Bi, boff, koff + 316u);
    mma_frag(f0, accR, accI);   // step 312
    mma_frag(f1, accR, accI);   // step 316
}

// Stage A:  T1[s] = sig[s] (planar, in d_out) * G   -> planar T1r/T1i in ws
__global__ void __launch_bounds__(256, 1)
k_gemmA(const float* __restrict__ sigR, const float* __restrict__ sigI,
        const float* __restrict__ Gr, const float* __restrict__ Gi,
        float* __restrict__ T1r, float* __restrict__ T1i)
{
    int wave = threadIdx.x >> 5;
    int lane = threadIdx.x & 31;
    int tile = blockIdx.x * 8 + wave;                    // 0..4999
    int s   = tile / TILES_PER_IMG;
    int rem = tile - s * TILES_PER_IMG;
    int m0 = (rem / NTILE16) * 64;
    int n0 = (rem % NTILE16) * 16;

    int ml   = lane & 15;
    unsigned koff = (unsigned)((lane >> 4) * 2);
    unsigned ncol = (unsigned)(n0 + ml);
    unsigned r0   = (unsigned)s * PIX + (unsigned)(m0 + ml) * HW320;

    v8f accR[4] = {{}, {}, {}, {}};
    v8f accI[4] = {{}, {}, {}, {}};
    cgemm_mainloop(sigR, sigI, r0, Gr, Gi, ncol, koff, accR, accI);

    unsigned col   = (unsigned)(n0 + (lane & 15));
    unsigned rhalf = (unsigned)((lane >> 4) * 8);
#pragma unroll
    for (int j = 0; j < 4; ++j) {
        unsigned base = (unsigned)s * PIX + ((unsigned)m0 + 16u * j + rhalf) * HW320 + col;
#pragma unroll
        for (int v = 0; v < 8; ++v) {
            T1r[base + (unsigned)v * HW320] = accR[j][v];
            T1i[base + (unsigned)v * HW320] = accI[j][v];
        }
    }
}

// Stage B:  out[s] = (1/320) * G * T1[s]   -> interleaved complex64 into d_out
__global__ void __launch_bounds__(256, 1)
k_gemmB(const float* __restrict__ Gr, const float* __restrict__ Gi,
        const float* __restrict__ T1r, const float* __restrict__ T1i,
        float* __restrict__ outc)
{
    int wave = threadIdx.x >> 5;
    int lane = threadIdx.x & 31;
    int tile = blockIdx.x * 8 + wave;
    int s   = tile / TILES_PER_IMG;
    int rem = tile - s * TILES_PER_IMG;
    int m0 = (rem / NTILE16) * 64;
    int n0 = (rem % NTILE16) * 16;

    int ml   = lane & 15;
    unsigned koff = (unsigned)((lane >> 4) * 2);
    unsigned r0   = (unsigned)(m0 + ml) * HW320;         // G rows contiguous in K
    unsigned boff = (unsigned)s * PIX + (unsigned)(n0 + ml);

    v8f accR[4] = {{}, {}, {}, {}};
    v8f accI[4] = {{}, {}, {}, {}};
    cgemm_mainloop(Gr, Gi, r0, T1r, T1i, boff, koff, accR, accI);

    const float sc = 1.0f / 320.0f;                      // fft2 norm="ortho" over 320x320
    unsigned col   = (unsigned)(n0 + (lane & 15));
    unsigned rhalf = (unsigned)((lane >> 4) * 8);
#pragma unroll
    for (int j = 0; j < 4; ++j) {
        unsigned ob = ((unsigned)s * PIX + ((unsigned)m0 + 16u * j + rhalf) * HW320 + col) * 2u;
#pragma unroll
        for (int v = 0; v < 8; ++v) {
            outc[ob + (unsigned)v * (2 * HW320)]     = accR[j][v] * sc;
            outc[ob + (unsigned)v * (2 * HW320) + 1] = accI[j][v] * sc;
        }
    }
}

// ---------------------------------------------------------------------------
// Launch.  Inputs: x[2,320,320,2] f32, aifci[601] f32, t_samp[601] f32,
//                  sample_time[50] f32.  Output: complex64 [50,320,320]
//                  = 10,240,000 interleaved floats.
// ws layout (floats): Gr[102400] | Gi[102400] | cK[64] | cA[64]
//                     | T1r[5,120,000] | T1i[5,120,000]   (~41.8 MB)
// d_out doubles as PLANAR staging for sig (real plane | imag plane), then is
// overwritten with the final interleaved complex64 by stage B.
// ---------------------------------------------------------------------------
extern "C" void kernel_launch(void* const* d_in, const int* in_sizes, int n_in,
                              void* d_out, int out_size, void* d_ws, size_t ws_size,
                              hipStream_t stream) {
    const float* x      = (const float*)d_in[0];
    const float* aifci  = (const float*)d_in[1];
    const float* t_samp = (const float*)d_in[2];
    const float* stime  = (const float*)d_in[3];
    int T     = in_sizes[1];
    int nsamp = in_sizes[3];

    float* out  = (float*)d_out;
    float* sigR = out;
    float* sigI = out + (size_t)NSAMP * PIX;
    float* ws   = (float*)d_ws;
    float* Gr   = ws;
    float* Gi   = ws + 102400;
    float* cK   = ws + 204800;
    float* cA   = ws + 204864;
    float* T1r  = ws + 204928;
    float* T1i  = T1r + (size_t)NSAMP * PIX;

    k_coef  <<<1,    64, 0, stream>>>(aifci, t_samp, stime, T, nsamp, cK, cA);
    k_dftmat<<<400, 256, 0, stream>>>(Gr, Gi);
    k_sig   <<<NSAMP * PIX / 256, 256, 0, stream>>>(x, cK, cA, sigR, sigI);
    k_gemmA <<<TOTAL_TILES / 8, 256, 0, stream>>>(sigR, sigI, Gr, Gi, T1r, T1i);
    k_gemmB <<<TOTAL_TILES / 8, 256, 0, stream>>>(Gr, Gi, T1r, T1i, out);
}